// SCA_base_43791486550172
// MI455X (gfx1250) — compile-verified
//
#include <hip/hip_runtime.h>
#include <cmath>

#define B_N     16
#define C_N     512
#define G_N     4
#define CG_N    128
#define R_N     16
#define HW_N    12544
#define HW4_N   3136
#define TOPK_N  384
#define TAU_F   0.5f
#define ALPHA_F 0.1f
#define EPS_F   1e-8f

typedef __attribute__((ext_vector_type(2))) float v2f;
typedef __attribute__((ext_vector_type(8))) float v8f;

// ---------------------------------------------------------------------------
// Kernel 1: fused avg+max pooling over HxW per (b,c).  One block per (b,c),
// float4 streaming loads, wave32 shuffle reduction then 8-wave LDS reduction.
// ---------------------------------------------------------------------------
__global__ __launch_bounds__(256) void pool_kernel(const float* __restrict__ x,
                                                   float* __restrict__ avg_out,
                                                   float* __restrict__ max_out) {
    const int bc = blockIdx.x;
    const float4* xp = (const float4*)(x + (size_t)bc * HW_N);

    float s = 0.0f;
    float m = -3.402823466e38f;
    for (int i = threadIdx.x; i < HW4_N; i += 256) {
        float4 v = xp[i];
        s += (v.x + v.y) + (v.z + v.w);
        m = fmaxf(m, fmaxf(fmaxf(v.x, v.y), fmaxf(v.z, v.w)));
    }
    // wave32 reduction
    #pragma unroll
    for (int off = 16; off > 0; off >>= 1) {
        s += __shfl_xor(s, off, 32);
        m = fmaxf(m, __shfl_xor(m, off, 32));
    }
    __shared__ float ssum[8];
    __shared__ float smax[8];
    const int wave = threadIdx.x >> 5;
    const int lane = threadIdx.x & 31;
    if (lane == 0) { ssum[wave] = s; smax[wave] = m; }
    __syncthreads();
    if (threadIdx.x == 0) {
        float ts = 0.0f, tm = -3.402823466e38f;
        #pragma unroll
        for (int w = 0; w < 8; ++w) { ts += ssum[w]; tm = fmaxf(tm, smax[w]); }
        avg_out[bc] = ts * (1.0f / (float)HW_N);
        max_out[bc] = tm;
    }
}

// ---------------------------------------------------------------------------
// Kernel 2: grouped MLP via V_WMMA_F32_16X16X4_F32 (one wave per group),
// then exact top-k rank selection + gumbel-sigmoid blend.  Single workgroup,
// 128 threads = 4 waves, EXEC all-1s through the WMMA region.
// ---------------------------------------------------------------------------
__global__ __launch_bounds__(128) void mlp_blend_kernel(
        const float* __restrict__ avg_in, const float* __restrict__ max_in,
        const float* __restrict__ W1, const float* __restrict__ b1,
        const float* __restrict__ W2, const float* __restrict__ b2,
        const float* __restrict__ noise_u, float* __restrict__ blend) {
    __shared__ float hLDS[G_N][16][16];      // relu(h) per group, D->A relayout
    __shared__ float scoreLDS[B_N][C_N];     // full score matrix

    const int tid  = threadIdx.x;
    const int g    = tid >> 5;               // group = wave id
    const int lane = tid & 31;
    const int mn   = lane & 15;              // A row (batch) / B,C,D column
    const int hi   = lane >> 4;              // half-wave select
    const int koff = hi * 2;                 // K sub-offset inside 16x4 step

    // gf[b, g, i] = combined[b, g*256 + i], combined = [avg(512) | max(512)]
    const float* src   = (g < 2) ? avg_in : max_in;
    const int    chbase = (g & 1) * 256;

    // ---- GEMM1: gf[16x256] x W1[g][256x16] -> acc[16x16] (K in steps of 4)
    v8f acc = {0.f, 0.f, 0.f, 0.f, 0.f, 0.f, 0.f, 0.f};
    for (int kb = 0; kb < 64; ++kb) {
        const int k = kb * 4 + koff;
        v2f a, b;
        a.x = src[mn * C_N + chbase + k];
        a.y = src[mn * C_N + chbase + k + 1];
        b.x = W1[(g * 256 + k) * R_N + mn];
        b.y = W1[(g * 256 + k + 1) * R_N + mn];
        acc = __builtin_amdgcn_wmma_f32_16x16x4_f32(false, a, false, b,
                                                    (short)0, acc, false, false);
    }
    // bias + relu, stash h in LDS in row-major (same-wave DS, no barrier needed)
    {
        const float bb = b1[g * R_N + mn];
        #pragma unroll
        for (int v = 0; v < 8; ++v)
            hLDS[g][v + 8 * hi][mn] = fmaxf(acc[v] + bb, 0.0f);
    }

    // ---- GEMM2: h[16x16] x W2[g][16x128] in 8 N-tiles of 16
    for (int nt = 0; nt < 8; ++nt) {
        v8f acc2 = {0.f, 0.f, 0.f, 0.f, 0.f, 0.f, 0.f, 0.f};
        #pragma unroll
        for (int kb = 0; kb < 4; ++kb) {
            const int k = kb * 4 + koff;
            v2f a, b;
            a.x = hLDS[g][mn][k];
            a.y = hLDS[g][mn][k + 1];
            b.x = W2[(g * R_N + k) * CG_N + nt * 16 + mn];
            b.y = W2[(g * R_N + k + 1) * CG_N + nt * 16 + mn];
            acc2 = __builtin_amdgcn_wmma_f32_16x16x4_f32(false, a, false, b,
                                                         (short)0, acc2, false, false);
        }
        const float bb2 = b2[g * CG_N + nt * 16 + mn];
        #pragma unroll
        for (int v = 0; v < 8; ++v)
            scoreLDS[v + 8 * hi][g * CG_N + nt * 16 + mn] = acc2[v] + bb2;
    }
    __syncthreads();

    // ---- exact top-k (rank count, index tie-break == jax.lax.top_k) + blend
    for (int e = tid; e < B_N * C_N; e += 128) {
        const int row = e >> 9;          // / 512
        const int col = e & (C_N - 1);
        const float sv = scoreLDS[row][col];
        int rank = 0;
        for (int j = 0; j < C_N; ++j) {
            const float o = scoreLDS[row][j];
            rank += (o > sv) ? 1 : ((o == sv && j < col) ? 1 : 0);
        }
        const float hard = (rank < TOPK_N) ? 1.0f : 0.0f;
        const float u    = noise_u[e];
        const float gum  = -logf(-logf(u + EPS_F) + EPS_F);
        const float soft = 1.0f / (1.0f + expf(-(sv + gum) * (1.0f / TAU_F)));
        // blend = mask + (1-mask)*alpha,  mask = hard*soft
        blend[e] = ALPHA_F + hard * soft * (1.0f - ALPHA_F);
    }
}

// ---------------------------------------------------------------------------
// Kernel 3: out = x * blend[b,c].  One block per (b,c): blend is a uniform
// scalar load; float4 streaming read + write.
// ---------------------------------------------------------------------------
__global__ __launch_bounds__(256) void scale_kernel(const float* __restrict__ x,
                                                    const float* __restrict__ blend,
                                                    float* __restrict__ out) {
    const int bc = blockIdx.x;
    const float s = blend[bc];
    const float4* xp = (const float4*)(x + (size_t)bc * HW_N);
    float4*       op = (float4*)(out + (size_t)bc * HW_N);
    for (int i = threadIdx.x; i < HW4_N; i += 256) {
        float4 v = xp[i];
        v.x *= s; v.y *= s; v.z *= s; v.w *= s;
        op[i] = v;
    }
}

extern "C" void kernel_launch(void* const* d_in, const int* in_sizes, int n_in,
                              void* d_out, int out_size, void* d_ws, size_t ws_size,
                              hipStream_t stream) {
    const float* x  = (const float*)d_in[0];
    const float* W1 = (const float*)d_in[1];
    const float* b1 = (const float*)d_in[2];
    const float* W2 = (const float*)d_in[3];
    const float* b2 = (const float*)d_in[4];
    const float* nu = (const float*)d_in[5];
    float* out = (float*)d_out;

    float* avg = (float*)d_ws;            // [B*C]
    float* mx  = avg + B_N * C_N;         // [B*C]
    float* bl  = mx + B_N * C_N;          // [B*C]

    pool_kernel<<<B_N * C_N, 256, 0, stream>>>(x, avg, mx);
    mlp_blend_kernel<<<1, 128, 0, stream>>>(avg, mx, W1, b1, W2, b2, nu, bl);
    scale_kernel<<<B_N * C_N, 256, 0, stream>>>(x, bl, out);
}